// GCNConv_86277303042052
// MI455X (gfx1250) — compile-verified
//
#include <hip/hip_runtime.h>
#include <cstdint>

#define N_NODES 50000
#define N_EDGES 800000
#define DIM     128

typedef __attribute__((ext_vector_type(2))) float v2f;
typedef __attribute__((ext_vector_type(8))) float v8f;

// ---------------------------------------------------------------------------
// Kernel 1: init degree accumulators. Self-loop contributes (e=1)+1 = 2.0 to
// both sender-degree and receiver-degree of every node, so start there.
// ---------------------------------------------------------------------------
__global__ void __launch_bounds__(256) k_init_deg(float* __restrict__ ds,
                                                  float* __restrict__ dr) {
    int i = blockIdx.x * blockDim.x + threadIdx.x;
    if (i < N_NODES) { ds[i] = 2.0f; dr[i] = 2.0f; }
}

// ---------------------------------------------------------------------------
// Kernel 2: per-edge degree accumulation: deg += edge_weight + 1.
// unsafeAtomicAdd -> global_atomic_add_f32 (L2-resolved, no CAS loop).
// ---------------------------------------------------------------------------
__global__ void __launch_bounds__(256) k_edge_deg(const int*   __restrict__ snd,
                                                  const int*   __restrict__ rcv,
                                                  const float* __restrict__ ew,
                                                  float* __restrict__ ds,
                                                  float* __restrict__ dr) {
    int i = blockIdx.x * blockDim.x + threadIdx.x;
    if (i < N_EDGES) {
        float w = ew[i] + 1.0f;
        unsafeAtomicAdd(&ds[snd[i]], w);
        unsafeAtomicAdd(&dr[rcv[i]], w);
    }
}

// ---------------------------------------------------------------------------
// Kernel 3: y = (nodes @ W) * rsqrt(max(deg_s,1)) * rsqrt(max(deg_r,1))
// using V_WMMA_F32_16X16X4_F32. Also writes out = y (the self-loop message,
// weight 1.0), which doubles as the deterministic initialization of d_out.
//
// Block = 128 threads = 4 waves; each wave computes a 16-row x 128-col strip.
// W is staged to LDS packed as pair-of-K-rows so each B fragment is one
// ds_load_b64:  w_lds[pair*256 + col*2 + (k&1)] = W[k*128 + col].
//
// WMMA f32 16x16x4 operand layout (ISA 7.12.2):
//   A (16x4):  lane m (0-15): A[m][k0], A[m][k0+1];  lane m+16: A[m][k0+2..3]
//   B (4x16):  lane n (0-15): B[k0][n], B[k0+1][n];  lane n+16: B[k0+2..3][n]
//   D (16x16): vgpr v, lanes 0-15 -> D[v][n], lanes 16-31 -> D[v+8][n]
// ---------------------------------------------------------------------------
__global__ void __launch_bounds__(128) k_gemm_scale(const float* __restrict__ nodes,
                                                    const float* __restrict__ Wm,
                                                    const float* __restrict__ ds,
                                                    const float* __restrict__ dr,
                                                    float* __restrict__ y,
                                                    float* __restrict__ out) {
    __shared__ float w_lds[DIM * DIM];  // 64 KB of the 320 KB/WGP LDS

    const int tid = threadIdx.x;

    // Cooperative stage of W into packed-pair LDS layout.
    for (int idx = tid; idx < DIM * DIM; idx += 128) {
        int k = idx >> 7;          // row of W
        int c = idx & (DIM - 1);   // col of W
        w_lds[((k >> 1) << 8) + (c << 1) + (k & 1)] = Wm[idx];
    }
    __syncthreads();

    const int lane = tid & 31;
    const int wave = tid >> 5;
    const int strip = blockIdx.x * 4 + wave;   // 16-row strip index

    if (strip * 16 < N_NODES) {                // uniform per wave: EXEC all-1s
        const int row0 = strip * 16;
        const int m    = lane & 15;            // matrix row (A) / col (B)
        const int koff = (lane >> 4) << 1;     // 0 for lanes 0-15, 2 for 16-31

        const float* arow = nodes + (size_t)(row0 + m) * DIM;

        v8f acc[8] = {};                       // 8 N-tiles of 16 -> 128 cols

        #pragma unroll 4
        for (int k0 = 0; k0 < DIM; k0 += 4) {
            v2f a = *(const v2f*)(arow + k0 + koff);          // global_load_b64
            const float* bbase = w_lds + (((k0 + koff) >> 1) << 8);
            #pragma unroll
            for (int t = 0; t < 8; ++t) {
                v2f b = *(const v2f*)(bbase + ((t * 16 + m) << 1));  // ds_load_b64
                acc[t] = __builtin_amdgcn_wmma_f32_16x16x4_f32(
                    /*neg_a=*/false, a, /*neg_b=*/false, b,
                    /*c_mod=*/(short)0, acc[t],
                    /*reuse_a=*/false, /*reuse_b=*/false);
            }
        }

        // Writeback with symmetric normalization; out = y handles self-loop.
        const int roff = (lane >> 4) << 3;     // 0 or 8
        #pragma unroll
        for (int v = 0; v < 8; ++v) {
            const int row = row0 + v + roff;
            const float a1 = ds[row];
            const float a2 = dr[row];
            const float sc = rsqrtf(fmaxf(a1, 1.0f)) * rsqrtf(fmaxf(a2, 1.0f));
            float* yrow = y   + (size_t)row * DIM;
            float* orow = out + (size_t)row * DIM;
            #pragma unroll
            for (int t = 0; t < 8; ++t) {
                const float val = acc[t][v] * sc;
                yrow[t * 16 + m] = val;
                orow[t * 16 + m] = val;
            }
        }
    }
}

// ---------------------------------------------------------------------------
// Kernel 4: edge scatter. One wave per edge; each lane gathers a float4
// (global_load_b128, 512B coalesced per wave — L2-resident since y is 25.6MB
// << 192MB L2) and issues 4 fp32 atomic adds into the receiver row.
// ---------------------------------------------------------------------------
__global__ void __launch_bounds__(256) k_edge_scatter(const int*   __restrict__ snd,
                                                      const int*   __restrict__ rcv,
                                                      const float* __restrict__ ew,
                                                      const float* __restrict__ y,
                                                      float* out) {
    const int gtid = blockIdx.x * blockDim.x + threadIdx.x;
    const int e    = gtid >> 5;
    if (e >= N_EDGES) return;
    const int lane = gtid & 31;

    const int   s = snd[e];
    const int   r = rcv[e];
    const float w = ew[e];

    const float4 v = *(const float4*)(y + (size_t)s * DIM + lane * 4);
    float* dst = out + (size_t)r * DIM + lane * 4;
    unsafeAtomicAdd(dst + 0, v.x * w);
    unsafeAtomicAdd(dst + 1, v.y * w);
    unsafeAtomicAdd(dst + 2, v.z * w);
    unsafeAtomicAdd(dst + 3, v.w * w);
}

// ---------------------------------------------------------------------------
extern "C" void kernel_launch(void* const* d_in, const int* in_sizes, int n_in,
                              void* d_out, int out_size, void* d_ws, size_t ws_size,
                              hipStream_t stream) {
    const float* nodes     = (const float*)d_in[0];
    const int*   senders   = (const int*)  d_in[1];
    const int*   receivers = (const int*)  d_in[2];
    const float* edges     = (const float*)d_in[3];
    const float* Wm        = (const float*)d_in[4];
    float*       out       = (float*)d_out;

    // Workspace layout: y [N*D], deg_s [N], deg_r [N]  (~26 MB)
    float* y  = (float*)d_ws;
    float* ds = y  + (size_t)N_NODES * DIM;
    float* dr = ds + N_NODES;

    k_init_deg<<<(N_NODES + 255) / 256, 256, 0, stream>>>(ds, dr);
    k_edge_deg<<<(N_EDGES + 255) / 256, 256, 0, stream>>>(senders, receivers, edges, ds, dr);

    const int n_strips = (N_NODES + 15) / 16;          // 3125
    k_gemm_scale<<<(n_strips + 3) / 4, 128, 0, stream>>>(nodes, Wm, ds, dr, y, out);

    const long long scatter_threads = (long long)N_EDGES * 32;
    k_edge_scatter<<<(int)((scatter_threads + 255) / 256), 256, 0, stream>>>(
        senders, receivers, edges, y, out);
}